// AffineTransformationLayer_20323785245282
// MI455X (gfx1250) — compile-verified
//
#include <hip/hip_runtime.h>

// Problem constants (fixed by the reference).
#define B_ 16
#define H_ 512
#define W_ 512
#define C_ 16

typedef float v4f __attribute__((ext_vector_type(4)));

// One thread per output pixel: scalar coordinate/weight math done once,
// amortized over all 16 channels (4x b128 per sample, 4x NT b128 out).
// Block = 256 threads = 256 pixels. Blocks per batch = (512*512)/256 = 1024.
__global__ __launch_bounds__(256) void affine_sample_kernel(
    const float* __restrict__ inputs,   // (B,H,W,C) f32
    const float* __restrict__ affines,  // (B,2,3)   f32
    float* __restrict__ out)            // (B,H,W,C) f32
{
    const int b   = blockIdx.x >> 10;                          // batch (uniform)
    const int pix = (blockIdx.x & 1023) * 256 + threadIdx.x;   // pixel in batch
    const int i   = pix >> 9;                                  // row (W=512)
    const int j   = pix & 511;                                 // col

    // Uniform per-block affine coefficients -> scalar loads.
    const float* A = affines + b * 6;
    const float a00 = A[0], a01 = A[1], a02 = A[2];
    const float a10 = A[3], a11 = A[4], a12 = A[5];

    // linspace(-1, 1, N): step = 2/(N-1); H_ == W_ so one constant.
    const float step = 2.0f / (float)(W_ - 1);
    const float xx = -1.0f + step * (float)j;
    const float yy = -1.0f + step * (float)i;

    const float tx = a00 * xx + a01 * yy + a02;
    const float ty = a10 * xx + a11 * yy + a12;
    const float x  = (tx + 1.0f) * (0.5f * (float)W_);
    const float y  = (ty + 1.0f) * (0.5f * (float)H_);

    int xl = (int)floorf(x);
    xl = min(max(xl, 0), W_ - 1);
    const int xr = min(xl + 1, W_ - 1);
    int yu = (int)floorf(y);
    yu = min(max(yu, 0), H_ - 1);
    const int yb = min(yu + 1, H_ - 1);

    const float dxl = x - (float)xl;
    const float dxr = (float)xr - x;
    const float dyu = y - (float)yu;
    const float dyb = (float)yb - y;

    const float wlu = expf(-dxl * dyu);
    const float wlb = expf(-dxl * dyb);
    const float wru = expf(-dxr * dyu);
    const float wrb = expf(-dxr * dyb);

    const float rden = 1.0f / (wlu + wlb + wru + wrb);

    // Four sample base addresses; 16 channels = 4 x b128 each with
    // immediate offsets off a single per-sample address.
    const float* base = inputs + (size_t)b * (H_ * (size_t)W_ * C_);
    const int rowU = yu << 9;   // yu * W_
    const int rowB = yb << 9;   // yb * W_
    const v4f* plu = (const v4f*)(base + ((rowU + xl) << 4));
    const v4f* plb = (const v4f*)(base + ((rowB + xl) << 4));
    const v4f* pru = (const v4f*)(base + ((rowU + xr) << 4));
    const v4f* prb = (const v4f*)(base + ((rowB + xr) << 4));

    const size_t g = (size_t)b * (H_ * (size_t)W_) + (size_t)pix;
    v4f* op = (v4f*)out + g * 4;

#pragma unroll
    for (int q = 0; q < 4; ++q) {
        v4f r = plu[q] * wlu;
        r += plb[q] * wlb;
        r += pru[q] * wru;
        r += prb[q] * wrb;
        r *= rden;
        // Write-once 256 MiB stream: non-temporal so it doesn't evict the
        // gather-reused input from the 192 MB L2.
        __builtin_nontemporal_store(r, op + q);
    }
}

extern "C" void kernel_launch(void* const* d_in, const int* in_sizes, int n_in,
                              void* d_out, int out_size, void* d_ws, size_t ws_size,
                              hipStream_t stream) {
    (void)in_sizes; (void)n_in; (void)out_size; (void)d_ws; (void)ws_size;
    const float* inputs  = (const float*)d_in[0];
    const float* affines = (const float*)d_in[1];
    float* out = (float*)d_out;

    const int total_threads = B_ * H_ * W_;   // 1 thread per pixel
    const int block = 256;
    const int grid  = total_threads / block;  // 16384

    affine_sample_kernel<<<grid, block, 0, stream>>>(inputs, affines, out);
}